// DilatedResidualBlock_59579786330605
// MI455X (gfx1250) — compile-verified
//
#include <hip/hip_runtime.h>
#include <hip/hip_bf16.h>

typedef __attribute__((ext_vector_type(2))) float v2f;
typedef __attribute__((ext_vector_type(8))) float v8f;

#define BS    2
#define NPTS  65536
#define KNN   16
#define EPSV  1e-5f
#define SLOPE 0.2f
#define NW    8
#define GRID  512
#define TOTW  (GRID * NW)
#define PTOT  (BS * NPTS)

// workspace layout (float offsets)
#define ST1_SUM   0
#define ST1_SQ    16
#define ST2_SUM   64
#define ST2_SQ    80
#define STA1_SUM  128
#define STA1_SQ   144
#define STA2_SUM  192
#define STA2_SQ   224
#define AFF1_SC   256
#define AFF1_SH   272
#define AFF2_SC   288
#define AFF2_SH   304
#define AFFA1_SC  320
#define AFFA1_SH  336
#define AFFA2_SC  352
#define AFFA2_SH  384
#define Z1_OFF    512   // [2][16][65536] pre-BN attentive-pool-1 output, later f_agg1 in place

// D = A(16x4) * B(4x16) + C, fp32 WMMA
__device__ __forceinline__ v8f wmma4(v2f a, v2f b, v8f c) {
  return __builtin_amdgcn_wmma_f32_16x16x4_f32(false, a, false, b, (short)0, c, false, false);
}

__device__ __forceinline__ float red16_add(float v) {
  v += __shfl_xor(v, 1, 32); v += __shfl_xor(v, 2, 32);
  v += __shfl_xor(v, 4, 32); v += __shfl_xor(v, 8, 32);
  return v;
}
__device__ __forceinline__ float red16_max(float v) {
  v = fmaxf(v, __shfl_xor(v, 1, 32)); v = fmaxf(v, __shfl_xor(v, 2, 32));
  v = fmaxf(v, __shfl_xor(v, 4, 32)); v = fmaxf(v, __shfl_xor(v, 8, 32));
  return v;
}

__global__ void k_zero(float* ws) {
  int i = blockIdx.x * blockDim.x + threadIdx.x;
  if (i < 512) ws[i] = 0.f;
}

// ---------------- stats of y1 = conv1x1(f_xyz, w_mlp1) + b over (b,n,k) ----------------
__global__ __launch_bounds__(256) void k_stats1(const float* __restrict__ xyz,
                                                const int* __restrict__ nidx,
                                                const float* __restrict__ w1,
                                                const float* __restrict__ b1,
                                                float* __restrict__ ws) {
  __shared__ float sW1[16 * 12];
  __shared__ float sB1[16];
  __shared__ float sFx[NW][12 * 16];
  int t = threadIdx.x;
  for (int i = t; i < 192; i += 256) { int o = i / 12, c = i % 12; sW1[i] = (c < 10) ? w1[o * 10 + c] : 0.f; }
  if (t < 16) sB1[t] = b1[t];
  __syncthreads();
  int lane = t & 31, w = t >> 5, lid = lane & 15, half = lane >> 4;
  float* fx = sFx[w];
  float ss[8] = {0,0,0,0,0,0,0,0}, sq[8] = {0,0,0,0,0,0,0,0};
  int wg = blockIdx.x * NW + w;
  for (int p = wg; p < PTOT; p += TOTW) {
    int b = p >> 16, n = p & (NPTS - 1);
    const float* xb = xyz + b * 3 * NPTS;
    int idx = nidx[((b * NPTS + n) << 4) + lid];
    float cx = xb[n], cy = xb[NPTS + n], cz = xb[2 * NPTS + n];
    float nx = xb[idx], ny = xb[NPTS + idx], nz = xb[2 * NPTS + idx];
    float rx = cx - nx, ry = cy - ny, rz = cz - nz;
    float dis = sqrtf(rx * rx + ry * ry + rz * rz);
    if (half == 0) {
      fx[0 * 16 + lid] = dis; fx[16 + lid] = rx; fx[32 + lid] = ry; fx[48 + lid] = rz;
      fx[64 + lid] = cx; fx[80 + lid] = cy; fx[96 + lid] = cz;
      fx[112 + lid] = nx; fx[128 + lid] = ny; fx[144 + lid] = nz;
      fx[160 + lid] = 0.f; fx[176 + lid] = 0.f;
    }
    __syncthreads();
    v8f y = {};
#pragma unroll
    for (int kc = 0; kc < 3; kc++) {
      int kb = kc * 4 + 2 * half;
      v2f a, bb;
      a.x = sW1[lid * 12 + kb]; a.y = sW1[lid * 12 + kb + 1];
      bb.x = fx[kb * 16 + lid]; bb.y = fx[(kb + 1) * 16 + lid];
      y = wmma4(a, bb, y);
    }
#pragma unroll
    for (int j = 0; j < 8; j++) {
      float v = y[j] + sB1[8 * half + j];
      ss[j] += v; sq[j] += v * v;
    }
    __syncthreads();
  }
#pragma unroll
  for (int j = 0; j < 8; j++) {
    float v = red16_add(ss[j]);
    float q = red16_add(sq[j]);
    if (lid == 0) { atomicAdd(&ws[ST1_SUM + 8 * half + j], v); atomicAdd(&ws[ST1_SQ + 8 * half + j], q); }
  }
}

__global__ void k_fin1(const float* g, const float* be, float* ws) {
  int t = threadIdx.x;
  if (t < 16) {
    float cnt = (float)PTOT * KNN;
    float m = ws[ST1_SUM + t] / cnt;
    float v = ws[ST1_SQ + t] / cnt - m * m;
    float sc = g[t] * rsqrtf(v + EPSV);
    ws[AFF1_SC + t] = sc; ws[AFF1_SH + t] = be[t] - m * sc;
  }
}

// ---------------- main pass 1 ----------------
__global__ __launch_bounds__(256) void k_pass1(const float* __restrict__ xyz,
                                               const float* __restrict__ feat,
                                               const int* __restrict__ nidx,
                                               const float* __restrict__ w1, const float* __restrict__ b1,
                                               const float* __restrict__ wfc, const float* __restrict__ bfc,
                                               const float* __restrict__ wap, const float* __restrict__ bap,
                                               const float* __restrict__ w2, const float* __restrict__ b2,
                                               float* __restrict__ ws) {
  __shared__ float sW1[16 * 12], sW2[16 * 16], sWfc[32 * 32], sWap[16 * 32];
  __shared__ float sB1[16], sB2[16], sBfc[32], sBap[16], sSc1[16], sSh1[16];
  __shared__ float sFx[NW][12 * 16];
  __shared__ float sCC[NW][32 * 16];
  __shared__ float sAgg[NW][32];
  int t = threadIdx.x;
  for (int i = t; i < 192; i += 256) { int o = i / 12, c = i % 12; sW1[i] = (c < 10) ? w1[o * 10 + c] : 0.f; }
  for (int i = t; i < 256; i += 256) sW2[i] = w2[i];
  for (int i = t; i < 1024; i += 256) sWfc[i] = wfc[i];
  for (int i = t; i < 512; i += 256) sWap[i] = wap[i];
  if (t < 16) { sB1[t] = b1[t]; sB2[t] = b2[t]; sBap[t] = bap[t]; sSc1[t] = ws[AFF1_SC + t]; sSh1[t] = ws[AFF1_SH + t]; }
  if (t < 32) sBfc[t] = bfc[t];
  __syncthreads();
  int lane = t & 31, w = t >> 5, lid = lane & 15, half = lane >> 4;
  float* fx = sFx[w]; float* cc = sCC[w]; float* agg = sAgg[w];
  float s2s[8] = {0,0,0,0,0,0,0,0}, s2q[8] = {0,0,0,0,0,0,0,0};
  float aps = 0.f, apq = 0.f;
  int wg = blockIdx.x * NW + w;
  for (int p = wg; p < PTOT; p += TOTW) {
    int b = p >> 16, n = p & (NPTS - 1);
    const float* xb = xyz + b * 3 * NPTS;
    int idx = nidx[((b * NPTS + n) << 4) + lid];
    float cx = xb[n], cy = xb[NPTS + n], cz = xb[2 * NPTS + n];
    float nx = xb[idx], ny = xb[NPTS + idx], nz = xb[2 * NPTS + idx];
    float rx = cx - nx, ry = cy - ny, rz = cz - nz;
    float dis = sqrtf(rx * rx + ry * ry + rz * rz);
    if (half == 0) {
      fx[0 * 16 + lid] = dis; fx[16 + lid] = rx; fx[32 + lid] = ry; fx[48 + lid] = rz;
      fx[64 + lid] = cx; fx[80 + lid] = cy; fx[96 + lid] = cz;
      fx[112 + lid] = nx; fx[128 + lid] = ny; fx[144 + lid] = nz;
      fx[160 + lid] = 0.f; fx[176 + lid] = 0.f;
    }
    const float* fb = feat + b * 16 * NPTS;
#pragma unroll
    for (int j = 0; j < 8; j++) { int ch = 8 * half + j; cc[ch * 16 + lid] = fb[ch * NPTS + idx]; }
    __syncthreads();
    // conv1 (10->16) via 3 fp32 WMMAs; BN1 + LReLU -> cc rows 16..31
    v8f y1 = {};
#pragma unroll
    for (int kc = 0; kc < 3; kc++) {
      int kb = kc * 4 + 2 * half;
      v2f a, bb;
      a.x = sW1[lid * 12 + kb]; a.y = sW1[lid * 12 + kb + 1];
      bb.x = fx[kb * 16 + lid]; bb.y = fx[(kb + 1) * 16 + lid];
      y1 = wmma4(a, bb, y1);
    }
#pragma unroll
    for (int j = 0; j < 8; j++) {
      int ch = 8 * half + j;
      float v = y1[j] + sB1[ch];
      v = v * sSc1[ch] + sSh1[ch];
      v = v >= 0.f ? v : SLOPE * v;
      cc[(16 + ch) * 16 + lid] = v;
    }
    __syncthreads();
    // y2 = w_mlp2 @ f_xyz1 + b (stats only this pass)
    v8f y2 = {};
#pragma unroll
    for (int kc = 0; kc < 4; kc++) {
      int kb = kc * 4 + 2 * half;
      v2f a, bb;
      a.x = sW2[lid * 16 + kb]; a.y = sW2[lid * 16 + kb + 1];
      bb.x = cc[(16 + kb) * 16 + lid]; bb.y = cc[(16 + kb + 1) * 16 + lid];
      y2 = wmma4(a, bb, y2);
    }
#pragma unroll
    for (int j = 0; j < 8; j++) {
      float v = y2[j] + sB2[8 * half + j];
      s2s[j] += v; s2q[j] += v * v;
    }
    // fc1 attention (32x32) as two 16-row tiles, softmax over K per channel, weighted sum
#pragma unroll
    for (int rt = 0; rt < 2; rt++) {
      int r = rt * 16;
      v8f att = {};
#pragma unroll
      for (int kc = 0; kc < 8; kc++) {
        int kb = kc * 4 + 2 * half;
        v2f a, bb;
        a.x = sWfc[(r + lid) * 32 + kb]; a.y = sWfc[(r + lid) * 32 + kb + 1];
        bb.x = cc[kb * 16 + lid]; bb.y = cc[(kb + 1) * 16 + lid];
        att = wmma4(a, bb, att);
      }
#pragma unroll
      for (int j = 0; j < 8; j++) {
        int ch = r + 8 * half + j;
        float v = att[j] + sBfc[ch];
        float m = red16_max(v);
        float e = __expf(v - m);
        float s = red16_add(e);
        float sc = e / s;
        float prod = red16_add(sc * cc[ch * 16 + lid]);
        if (lid == 0) agg[ch] = prod;
      }
    }
    __syncthreads();
    if (lane < 16) {
      float z = sBap[lane];
#pragma unroll
      for (int i = 0; i < 32; i++) z += sWap[lane * 32 + i] * agg[i];
      aps += z; apq += z * z;
      ws[Z1_OFF + (b * 16 + lane) * NPTS + n] = z;
    }
    __syncthreads();
  }
#pragma unroll
  for (int j = 0; j < 8; j++) {
    float v = red16_add(s2s[j]);
    float q = red16_add(s2q[j]);
    if (lid == 0) { atomicAdd(&ws[ST2_SUM + 8 * half + j], v); atomicAdd(&ws[ST2_SQ + 8 * half + j], q); }
  }
  if (lane < 16) { atomicAdd(&ws[STA1_SUM + lane], aps); atomicAdd(&ws[STA1_SQ + lane], apq); }
}

__global__ void k_fin2(const float* g2, const float* be2, const float* ga1, const float* bea1, float* ws) {
  int t = threadIdx.x;
  if (t < 16) {
    float cnt = (float)PTOT * KNN;
    float m = ws[ST2_SUM + t] / cnt, v = ws[ST2_SQ + t] / cnt - m * m;
    float sc = g2[t] * rsqrtf(v + EPSV);
    ws[AFF2_SC + t] = sc; ws[AFF2_SH + t] = be2[t] - m * sc;
    float cnt2 = (float)PTOT;
    float m2 = ws[STA1_SUM + t] / cnt2, v2 = ws[STA1_SQ + t] / cnt2 - m2 * m2;
    float sc2 = ga1[t] * rsqrtf(v2 + EPSV);
    ws[AFFA1_SC + t] = sc2; ws[AFFA1_SH + t] = bea1[t] - m2 * sc2;
  }
}

__global__ void k_apply_a1(float* ws) {
  int i = blockIdx.x * blockDim.x + threadIdx.x;
  if (i < BS * 16 * NPTS) {
    int ch = (i >> 16) & 15;
    float v = ws[Z1_OFF + i];
    v = v * ws[AFFA1_SC + ch] + ws[AFFA1_SH + ch];
    ws[Z1_OFF + i] = v >= 0.f ? v : SLOPE * v;
  }
}

// ---------------- main pass 2 ----------------
__global__ __launch_bounds__(256) void k_pass2(const float* __restrict__ xyz,
                                               const int* __restrict__ nidx,
                                               const float* __restrict__ w1, const float* __restrict__ b1,
                                               const float* __restrict__ w2, const float* __restrict__ b2,
                                               const float* __restrict__ wfc, const float* __restrict__ bfc,
                                               const float* __restrict__ wap, const float* __restrict__ bap,
                                               float* __restrict__ ws, float* __restrict__ out) {
  __shared__ float sW1[16 * 12], sW2[16 * 16], sWfc[32 * 32], sWap[32 * 32];
  __shared__ float sB1[16], sB2[16], sBfc[32], sBap[32];
  __shared__ float sSc1[16], sSh1[16], sSc2[16], sSh2[16];
  __shared__ float sFx[NW][12 * 16];
  __shared__ float sCC[NW][32 * 16];
  __shared__ float sAgg[NW][32];
  int t = threadIdx.x;
  for (int i = t; i < 192; i += 256) { int o = i / 12, c = i % 12; sW1[i] = (c < 10) ? w1[o * 10 + c] : 0.f; }
  for (int i = t; i < 256; i += 256) sW2[i] = w2[i];
  for (int i = t; i < 1024; i += 256) { sWfc[i] = wfc[i]; sWap[i] = wap[i]; }
  if (t < 16) {
    sB1[t] = b1[t]; sB2[t] = b2[t];
    sSc1[t] = ws[AFF1_SC + t]; sSh1[t] = ws[AFF1_SH + t];
    sSc2[t] = ws[AFF2_SC + t]; sSh2[t] = ws[AFF2_SH + t];
  }
  if (t < 32) { sBfc[t] = bfc[t]; sBap[t] = bap[t]; }
  __syncthreads();
  int lane = t & 31, w = t >> 5, lid = lane & 15, half = lane >> 4;
  float* fx = sFx[w]; float* cc = sCC[w]; float* agg = sAgg[w];
  float aps = 0.f, apq = 0.f;  // ap2 stats, channel = lane
  int wg = blockIdx.x * NW + w;
  for (int p = wg; p < PTOT; p += TOTW) {
    int b = p >> 16, n = p & (NPTS - 1);
    const float* xb = xyz + b * 3 * NPTS;
    int idx = nidx[((b * NPTS + n) << 4) + lid];
    float cx = xb[n], cy = xb[NPTS + n], cz = xb[2 * NPTS + n];
    float nx = xb[idx], ny = xb[NPTS + idx], nz = xb[2 * NPTS + idx];
    float rx = cx - nx, ry = cy - ny, rz = cz - nz;
    float dis = sqrtf(rx * rx + ry * ry + rz * rz);
    if (half == 0) {
      fx[0 * 16 + lid] = dis; fx[16 + lid] = rx; fx[32 + lid] = ry; fx[48 + lid] = rz;
      fx[64 + lid] = cx; fx[80 + lid] = cy; fx[96 + lid] = cz;
      fx[112 + lid] = nx; fx[128 + lid] = ny; fx[144 + lid] = nz;
      fx[160 + lid] = 0.f; fx[176 + lid] = 0.f;
    }
    const float* fg = ws + Z1_OFF + b * 16 * NPTS;
#pragma unroll
    for (int j = 0; j < 8; j++) { int ch = 8 * half + j; cc[ch * 16 + lid] = fg[ch * NPTS + idx]; }
    __syncthreads();
    // rebuild f_xyz1
    v8f y1 = {};
#pragma unroll
    for (int kc = 0; kc < 3; kc++) {
      int kb = kc * 4 + 2 * half;
      v2f a, bb;
      a.x = sW1[lid * 12 + kb]; a.y = sW1[lid * 12 + kb + 1];
      bb.x = fx[kb * 16 + lid]; bb.y = fx[(kb + 1) * 16 + lid];
      y1 = wmma4(a, bb, y1);
    }
#pragma unroll
    for (int j = 0; j < 8; j++) {
      int ch = 8 * half + j;
      float v = y1[j] + sB1[ch];
      v = v * sSc1[ch] + sSh1[ch];
      v = v >= 0.f ? v : SLOPE * v;
      cc[(16 + ch) * 16 + lid] = v;
    }
    __syncthreads();
    // f_xyz2 = LReLU(BN2(w_mlp2 @ f_xyz1 + b2))
    v8f y2 = {};
#pragma unroll
    for (int kc = 0; kc < 4; kc++) {
      int kb = kc * 4 + 2 * half;
      v2f a, bb;
      a.x = sW2[lid * 16 + kb]; a.y = sW2[lid * 16 + kb + 1];
      bb.x = cc[(16 + kb) * 16 + lid]; bb.y = cc[(16 + kb + 1) * 16 + lid];
      y2 = wmma4(a, bb, y2);
    }
    float f2[8];
#pragma unroll
    for (int j = 0; j < 8; j++) {
      int ch = 8 * half + j;
      float v = y2[j] + sB2[ch];
      v = v * sSc2[ch] + sSh2[ch];
      f2[j] = v >= 0.f ? v : SLOPE * v;
    }
    __syncthreads();
#pragma unroll
    for (int j = 0; j < 8; j++) { int ch = 8 * half + j; cc[(16 + ch) * 16 + lid] = f2[j]; }
    __syncthreads();
    // fc2 attention + softmax + pooled sum
#pragma unroll
    for (int rt = 0; rt < 2; rt++) {
      int r = rt * 16;
      v8f att = {};
#pragma unroll
      for (int kc = 0; kc < 8; kc++) {
        int kb = kc * 4 + 2 * half;
        v2f a, bb;
        a.x = sWfc[(r + lid) * 32 + kb]; a.y = sWfc[(r + lid) * 32 + kb + 1];
        bb.x = cc[kb * 16 + lid]; bb.y = cc[(kb + 1) * 16 + lid];
        att = wmma4(a, bb, att);
      }
#pragma unroll
      for (int j = 0; j < 8; j++) {
        int ch = r + 8 * half + j;
        float v = att[j] + sBfc[ch];
        float m = red16_max(v);
        float e = __expf(v - m);
        float s = red16_add(e);
        float sc = e / s;
        float prod = red16_add(sc * cc[ch * 16 + lid]);
        if (lid == 0) agg[ch] = prod;
      }
    }
    __syncthreads();
    {
      float z = sBap[lane];
#pragma unroll
      for (int i = 0; i < 32; i++) z += sWap[lane * 32 + i] * agg[i];
      aps += z; apq += z * z;
      out[(b * 32 + lane) * NPTS + n] = z;
    }
    __syncthreads();
  }
  atomicAdd(&ws[STA2_SUM + lane], aps);
  atomicAdd(&ws[STA2_SQ + lane], apq);
}

__global__ void k_fin3(const float* g, const float* be, float* ws) {
  int t = threadIdx.x;
  if (t < 32) {
    float cnt = (float)PTOT;
    float m = ws[STA2_SUM + t] / cnt, v = ws[STA2_SQ + t] / cnt - m * m;
    float sc = g[t] * rsqrtf(v + EPSV);
    ws[AFFA2_SC + t] = sc; ws[AFFA2_SH + t] = be[t] - m * sc;
  }
}

__global__ void k_apply_out(float* out, const float* ws) {
  int i = blockIdx.x * blockDim.x + threadIdx.x;
  if (i < BS * 32 * NPTS) {
    int ch = (i >> 16) & 31;
    float v = out[i];
    v = v * ws[AFFA2_SC + ch] + ws[AFFA2_SH + ch];
    out[i] = v >= 0.f ? v : SLOPE * v;
  }
}

extern "C" void kernel_launch(void* const* d_in, const int* in_sizes, int n_in,
                              void* d_out, int out_size, void* d_ws, size_t ws_size,
                              hipStream_t stream) {
  const float* xyz    = (const float*)d_in[0];
  const float* feat   = (const float*)d_in[1];
  const int*   nidx   = (const int*)d_in[2];
  const float* w_mlp1 = (const float*)d_in[3];
  const float* b_mlp1 = (const float*)d_in[4];
  const float* g_mlp1 = (const float*)d_in[5];
  const float* be_mlp1= (const float*)d_in[6];
  const float* w_fc1  = (const float*)d_in[7];
  const float* b_fc1  = (const float*)d_in[8];
  const float* w_ap1  = (const float*)d_in[9];
  const float* b_ap1  = (const float*)d_in[10];
  const float* g_ap1  = (const float*)d_in[11];
  const float* be_ap1 = (const float*)d_in[12];
  const float* w_mlp2 = (const float*)d_in[13];
  const float* b_mlp2 = (const float*)d_in[14];
  const float* g_mlp2 = (const float*)d_in[15];
  const float* be_mlp2= (const float*)d_in[16];
  const float* w_fc2  = (const float*)d_in[17];
  const float* b_fc2  = (const float*)d_in[18];
  const float* w_ap2  = (const float*)d_in[19];
  const float* b_ap2  = (const float*)d_in[20];
  const float* g_ap2  = (const float*)d_in[21];
  const float* be_ap2 = (const float*)d_in[22];
  float* out = (float*)d_out;
  float* ws  = (float*)d_ws;
  (void)in_sizes; (void)n_in; (void)out_size; (void)ws_size;

  k_zero<<<dim3(2), dim3(256), 0, stream>>>(ws);
  k_stats1<<<dim3(GRID), dim3(256), 0, stream>>>(xyz, nidx, w_mlp1, b_mlp1, ws);
  k_fin1<<<dim3(1), dim3(32), 0, stream>>>(g_mlp1, be_mlp1, ws);
  k_pass1<<<dim3(GRID), dim3(256), 0, stream>>>(xyz, feat, nidx, w_mlp1, b_mlp1,
                                                w_fc1, b_fc1, w_ap1, b_ap1, w_mlp2, b_mlp2, ws);
  k_fin2<<<dim3(1), dim3(32), 0, stream>>>(g_mlp2, be_mlp2, g_ap1, be_ap1, ws);
  k_apply_a1<<<dim3((BS * 16 * NPTS + 255) / 256), dim3(256), 0, stream>>>(ws);
  k_pass2<<<dim3(GRID), dim3(256), 0, stream>>>(xyz, nidx, w_mlp1, b_mlp1, w_mlp2, b_mlp2,
                                                w_fc2, b_fc2, w_ap2, b_ap2, ws, out);
  k_fin3<<<dim3(1), dim3(32), 0, stream>>>(g_ap2, be_ap2, ws);
  k_apply_out<<<dim3((BS * 32 * NPTS + 255) / 256), dim3(256), 0, stream>>>(out, ws);
}